// Dynamic_Loss_78683800863374
// MI455X (gfx1250) — compile-verified
//
#include <hip/hip_runtime.h>
#include <hip/hip_bf16.h>
#include <stdint.h>

// ---------------------------------------------------------------------------
// Dynamic segmentation loss, fused 2-pass pipeline for MI455X (gfx1250).
// Bandwidth-bound: ~344MB mandatory fp32 reads -> ~15us floor @ 23.3 TB/s.
// Pass 1: async-stage [20ch logit + 20ch ema] x 256 pixels into LDS per block
//         (CDNA5 global_load_async_to_lds_b128), compute lse/argmax/scores,
//         LDS-atomic per-class sums/counts, persist per-pixel (-logp, dyn).
// Tiny kernel: per-class EMA update -> loss weights.
// Pass 2: light re-read of the 10.5MB per-pixel arrays -> weighted masked mean.
// ---------------------------------------------------------------------------

namespace {
constexpr int kNC     = 20;        // total classes
constexpr int kPast   = 15;        // sum(num_classes[:-1])
constexpr int kIgnore = 255;
constexpr int kHW     = 512 * 512; // 262144 = 2^18
constexpr int kNPix   = 8 * kHW;   // 2097152
constexpr int kBlk    = 256;
}

// --- CDNA5 async global->LDS staging (guarded: falls back to direct loads) --
#if defined(__HIP_DEVICE_COMPILE__) && __has_builtin(__builtin_amdgcn_global_load_async_to_lds_b128)
#define DL_HAVE_ASYNC 1
#else
#define DL_HAVE_ASYNC 0
#endif

#if DL_HAVE_ASYNC
constexpr int kStageFloats = 2 * kNC * kBlk;   // 40 KB of LDS per block
// Builtin parameter types (from hipcc diagnostic): v4i in AS1 / AS3.
typedef int dl_v4i __attribute__((vector_size(4 * sizeof(int))));
typedef __attribute__((address_space(1))) dl_v4i* dl_gptr;
typedef __attribute__((address_space(3))) dl_v4i* dl_lptr;
#else
constexpr int kStageFloats = 1;
#endif

__device__ __forceinline__ void dl_stage_issue(const float* __restrict__ glb,
                                               const float* __restrict__ geb,
                                               float* s_stage) {
#if DL_HAVE_ASYNC
  const int tid = threadIdx.x;
  // Each thread async-copies 16B per issue; 256 threads x 16B = 4KB/issue.
  // 20KB per tensor -> 5 issues per tensor per thread. Addresses 16B-aligned.
#pragma unroll
  for (int it = 0; it < 5; ++it) {
    const int o = it * 4096 + tid * 16;   // byte offset inside the 20KB region
    const int c = o >> 10;                // channel (1KB per channel row)
    const int w = o & 1023;               // byte within the channel row
    const char* g0 = reinterpret_cast<const char*>(glb + (size_t)c * kHW) + w;
    const char* g1 = reinterpret_cast<const char*>(geb + (size_t)c * kHW) + w;
    char* l0 = reinterpret_cast<char*>(s_stage) + o;
    char* l1 = reinterpret_cast<char*>(s_stage) + kNC * kBlk * 4 + o;
    __builtin_amdgcn_global_load_async_to_lds_b128((dl_gptr)g0, (dl_lptr)l0,
                                                   0, 0);
    __builtin_amdgcn_global_load_async_to_lds_b128((dl_gptr)g1, (dl_lptr)l1,
                                                   0, 0);
  }
#else
  (void)glb; (void)geb; (void)s_stage;
#endif
}

__device__ __forceinline__ void dl_stage_wait() {
#if DL_HAVE_ASYNC
#if __has_builtin(__builtin_amdgcn_s_wait_asynccnt)
  __builtin_amdgcn_s_wait_asynccnt(0);
#else
  asm volatile("s_wait_asynccnt 0" ::: "memory");
#endif
#endif
}

// Per-pixel math shared by pass1 and the recompute fallback.
// gl/ge are per-thread base pointers (stride kHW between channels) for the
// direct-load fallback; s_stage holds the LDS-staged tiles for the async path.
__device__ __forceinline__ void dl_compute_pixel(const float* __restrict__ gl,
                                                 const float* __restrict__ ge,
                                                 const float* s_stage,
                                                 const float* __restrict__ thresh,
                                                 int real,
                                                 int& label, float& score,
                                                 int& dyn, float& nlp) {
  const int tid = threadIdx.x;
#if DL_HAVE_ASYNC
#define DL_L(c) s_stage[(c) * kBlk + tid]
#define DL_E(c) s_stage[(kNC + (c)) * kBlk + tid]
  (void)gl; (void)ge;
#else
#define DL_L(c) gl[(size_t)(c) * kHW]
#define DL_E(c) ge[(size_t)(c) * kHW]
  (void)s_stage; (void)tid;
#endif
  // ---- logit side: max over all 20, argmax over first 15 (first-max wins) --
  float m = -3.402823466e38f, best = -3.402823466e38f;
  int am = 0;
#pragma unroll
  for (int c = 0; c < kNC; ++c) {
    const float x = DL_L(c);
    m = fmaxf(m, x);
    if (c < kPast && x > best) { best = x; am = c; }
  }
  float se = 0.f;
#pragma unroll
  for (int c = 0; c < kNC; ++c) se += __expf(DL_L(c) - m);
  const float lse = m + __logf(se);

  label = (real >= kPast) ? real : am;                  // 0..19 or 255
  const int idx = (real == kIgnore) ? 0 : label;        // always 0..19
  score = __expf(DL_L(idx) - lse);                      // softmax prob at idx

  // ---- EMA-teacher side -------------------------------------------------
  float ebest = -3.402823466e38f;
  int eam = 0;
#pragma unroll
  for (int c = 0; c < kPast; ++c) {
    const float x = DL_E(c);
    if (x > ebest) { ebest = x; eam = c; }
  }
  const int elab = (real >= kPast) ? real : eam;        // 0..19 or 255
  const int eidx = (real == kIgnore) ? 0 : elab;        // always 0..19
  const float esc = DL_E(eidx);
  const bool vcls = elab < kNC;
  const float thr = thresh[vcls ? elab : (kNC - 1)];
  const bool mask = vcls && (esc >= thr);
  dyn = mask ? elab : kIgnore;

  const int cidx = (dyn < kNC) ? dyn : (kNC - 1);
  nlp = lse - DL_L(cidx);                               // -log_softmax[cidx]
#undef DL_L
#undef DL_E
}

__device__ __forceinline__ void dl_wave_accum(float num, float den,
                                              float* g_num, float* g_den) {
#pragma unroll
  for (int off = 16; off > 0; off >>= 1) {   // wave32 tree reduction
    num += __shfl_down(num, off, 32);
    den += __shfl_down(den, off, 32);
  }
  if ((threadIdx.x & 31) == 0) {
    atomicAdd(g_num, num);
    atomicAdd(g_den, den);
  }
}

// ---- Pass 1: per-pixel stats + per-class segment sums ----------------------
template <bool STORE>
__global__ __launch_bounds__(kBlk) void dl_pass1(
    const float* __restrict__ logit, const float* __restrict__ ema,
    const float* __restrict__ thresh, const int* __restrict__ real_labels,
    float* __restrict__ g_sum, float* __restrict__ g_cnt,
    float* __restrict__ nlp_out, unsigned char* __restrict__ dyn_out) {
  __shared__ __align__(16) float s_stage[kStageFloats];
  __shared__ float s_sum[kNC];
  __shared__ float s_cnt[kNC];

  const int tid = threadIdx.x;
  const size_t p0 = (size_t)blockIdx.x * kBlk;      // never crosses a batch
  const int b = (int)(p0 >> 18);
  const size_t s0 = p0 & (size_t)(kHW - 1);
  const float* glb = logit + (size_t)b * kNC * kHW + s0;
  const float* geb = ema   + (size_t)b * kNC * kHW + s0;

  dl_stage_issue(glb, geb, s_stage);
  if (tid < kNC) { s_sum[tid] = 0.f; s_cnt[tid] = 0.f; }
  dl_stage_wait();
  __syncthreads();

  const int real = real_labels[p0 + tid];
  int label, dyn;
  float score, nlp;
  dl_compute_pixel(glb + tid, geb + tid, s_stage, thresh, real,
                   label, score, dyn, nlp);

  if (label < kNC) {                 // ignore bucket (255) is dropped
    atomicAdd(&s_sum[label], score); // ds_add_f32
    atomicAdd(&s_cnt[label], 1.0f);
  }
  if (STORE) {
    nlp_out[p0 + tid] = nlp;
    dyn_out[p0 + tid] = (unsigned char)dyn;
  }
  __syncthreads();
  if (tid < kNC && s_cnt[tid] > 0.f) {
    atomicAdd(&g_sum[tid], s_sum[tid]);
    atomicAdd(&g_cnt[tid], s_cnt[tid]);
  }
}

// ---- Tiny per-class EMA update -> loss weights -----------------------------
__global__ void dl_finalize_mpa(const float* __restrict__ g_sum,
                                const float* __restrict__ g_cnt,
                                const float* __restrict__ mpa_in,
                                float* __restrict__ wmult) {
  const int c = threadIdx.x;
  if (c < kNC) {
    const float cnt = g_cnt[c];
    const float mean = g_sum[c] / fmaxf(cnt, 1.0f);
    const float mpa = mpa_in[c];
    const float nm = (cnt > 0.f)
                         ? ((mpa == -1.0f) ? mean : (0.01f * mean + 0.99f * mpa))
                         : mpa;
    wmult[c] = 1.0f - nm;
  }
}

// ---- Pass 2 (light): weighted masked mean from persisted per-pixel data ----
__global__ __launch_bounds__(kBlk) void dl_pass3_light(
    const float* __restrict__ nlp, const unsigned char* __restrict__ dyn,
    const float* __restrict__ wmult, float* __restrict__ g_num,
    float* __restrict__ g_den) {
  __shared__ float s_w[kNC];
  if (threadIdx.x < kNC) s_w[threadIdx.x] = wmult[threadIdx.x];
  __syncthreads();
  const size_t stride = (size_t)gridDim.x * blockDim.x;
  float num = 0.f, den = 0.f;
  for (size_t p = (size_t)blockIdx.x * blockDim.x + threadIdx.x; p < kNPix;
       p += stride) {
    __builtin_prefetch(&nlp[p + stride], 0, 0);   // global_prefetch_b8
    const unsigned char d = dyn[p];
    if (d != (unsigned char)kIgnore) {
      num += nlp[p] * s_w[d];
      den += 1.f;
    }
  }
  dl_wave_accum(num, den, g_num, g_den);
}

// ---- Pass 2 (fallback): recompute if workspace too small -------------------
__global__ __launch_bounds__(kBlk) void dl_pass3_recompute(
    const float* __restrict__ logit, const float* __restrict__ ema,
    const float* __restrict__ thresh, const int* __restrict__ real_labels,
    const float* __restrict__ wmult, float* __restrict__ g_num,
    float* __restrict__ g_den) {
  __shared__ __align__(16) float s_stage[kStageFloats];
  __shared__ float s_w[kNC];
  const int tid = threadIdx.x;
  const size_t p0 = (size_t)blockIdx.x * kBlk;
  const int b = (int)(p0 >> 18);
  const size_t s0 = p0 & (size_t)(kHW - 1);
  const float* glb = logit + (size_t)b * kNC * kHW + s0;
  const float* geb = ema   + (size_t)b * kNC * kHW + s0;

  dl_stage_issue(glb, geb, s_stage);
  if (tid < kNC) s_w[tid] = wmult[tid];
  dl_stage_wait();
  __syncthreads();

  const int real = real_labels[p0 + tid];
  int label, dyn;
  float score, nlp;
  dl_compute_pixel(glb + tid, geb + tid, s_stage, thresh, real,
                   label, score, dyn, nlp);
  float num = 0.f, den = 0.f;
  if (dyn != kIgnore) { num = nlp * s_w[dyn]; den = 1.f; }
  dl_wave_accum(num, den, g_num, g_den);
}

__global__ void dl_finalize_out(const float* __restrict__ g_num,
                                const float* __restrict__ g_den,
                                float* __restrict__ out) {
  out[0] = g_num[0] / g_den[0];
}

// ---------------------------------------------------------------------------
extern "C" void kernel_launch(void* const* d_in, const int* in_sizes, int n_in,
                              void* d_out, int out_size, void* d_ws,
                              size_t ws_size, hipStream_t stream) {
  const float* logit  = (const float*)d_in[0];   // [8,20,512,512] f32
  const float* ema    = (const float*)d_in[1];   // [8,20,512,512] f32
  const float* thresh = (const float*)d_in[2];   // [20] f32
  const int* real_lab = (const int*)d_in[3];     // [8,512,512] i32
  const float* mpa    = (const float*)d_in[4];   // [20] f32
  float* out = (float*)d_out;                    // scalar f32

  // Workspace layout: [nlp f32 x 2M | dyn u8 x 2M | 62 f32 scalars]
  const size_t big = (size_t)kNPix * 4 + (size_t)kNPix;  // 10,485,760 B
  const bool store = ws_size >= big + 1024;
  char* wsb = (char*)d_ws;
  float* nlp_arr = (float*)wsb;
  unsigned char* dyn_arr = (unsigned char*)(wsb + (size_t)kNPix * 4);
  char* small = store ? (wsb + big) : wsb;
  float* g_sum = (float*)small;          // [20]
  float* g_cnt = g_sum + kNC;            // [20]
  float* g_w   = g_sum + 2 * kNC;        // [20]
  float* g_num = g_sum + 3 * kNC;        // [1]
  float* g_den = g_num + 1;              // [1]

  (void)hipMemsetAsync(small, 0, (3 * kNC + 2) * sizeof(float), stream);

  const int nblk = kNPix / kBlk;  // 8192
  if (store) {
    dl_pass1<true><<<nblk, kBlk, 0, stream>>>(logit, ema, thresh, real_lab,
                                              g_sum, g_cnt, nlp_arr, dyn_arr);
  } else {
    dl_pass1<false><<<nblk, kBlk, 0, stream>>>(logit, ema, thresh, real_lab,
                                               g_sum, g_cnt, nullptr, nullptr);
  }
  dl_finalize_mpa<<<1, 32, 0, stream>>>(g_sum, g_cnt, mpa, g_w);
  if (store) {
    dl_pass3_light<<<1024, kBlk, 0, stream>>>(nlp_arr, dyn_arr, g_w, g_num,
                                              g_den);
  } else {
    dl_pass3_recompute<<<nblk, kBlk, 0, stream>>>(logit, ema, thresh, real_lab,
                                                  g_w, g_num, g_den);
  }
  dl_finalize_out<<<1, 1, 0, stream>>>(g_num, g_den, out);
}